// SpaAggregator_38766374814283
// MI455X (gfx1250) — compile-verified
//
#include <hip/hip_runtime.h>
#include <stdint.h>

// ---------------------------------------------------------------------------
// SpaAggregator gather-mean:  out[b,:] = (1/K) * sum_j table[idx[b,j], :]
//   B=50000, K=32, D=128, fp32.
//
// MI455X roofline: table = 200000*128*4 = 102.4 MB -> resident in 192 MB L2.
// Gather traffic = 819 MB of L2 hits; HBM sees ~141 MB total (~6 us at
// 23.3 TB/s). Bound by L2 gather bandwidth / latency hiding, not compute.
// WMMA cannot express per-row gathers (B operand is shared across A rows),
// so the CDNA5 feature that fits is the ASYNCcnt-tracked
// GLOBAL_LOAD_ASYNC_TO_LDS_B128 software pipeline: one wave32 owns one node,
// each async op moves one full 512B embedding row (16B/lane, coalesced),
// 8 rows in flight per wave, consumed from LDS via ds_load_b128.
// NT hints keep the write-once output / read-once indices from evicting the
// L2-resident table.
// ---------------------------------------------------------------------------

#define WAVES_PER_BLOCK 8
#define THREADS_PER_BLOCK (WAVES_PER_BLOCK * 32)
#define DEPTH 8            // async rows in flight per wave
#define DIM 128            // feature dim (reference)
#define ROW_BYTES (DIM * 4)

typedef __attribute__((ext_vector_type(4))) float v4f;

// GV form: per-lane 64-bit global address.
__device__ __forceinline__ void async_b128_gv(unsigned lds_off, const float* gsrc) {
  asm volatile("global_load_async_to_lds_b128 %0, %1, off"
               :: "v"(lds_off), "v"(gsrc) : "memory");
}
// GVS form: scalar 64-bit base (SGPR pair) + 32-bit per-lane offset.
__device__ __forceinline__ void async_b128_gvs(unsigned lds_off, unsigned voff,
                                               uint64_t sbase) {
  asm volatile("global_load_async_to_lds_b128 %0, %1, %2"
               :: "v"(lds_off), "v"(voff), "s"(sbase) : "memory");
}
template <int N>
__device__ __forceinline__ void wait_asynccnt() {
  asm volatile("s_wait_asynccnt %0" :: "n"(N) : "memory");
}
__device__ __forceinline__ void wait_dscnt0() {
  asm volatile("s_wait_dscnt 0" ::: "memory");
}

__global__ __launch_bounds__(THREADS_PER_BLOCK)
void SpaAggregator_gather_mean_kernel(const float* __restrict__ table,
                                      const int* __restrict__ neigh_idx,
                                      float* __restrict__ out,
                                      int B, int K) {
  // 8 waves * 8 slots * 512B = 32 KB per block (WGP has 320 KB).
  __shared__ __align__(16) float smem[WAVES_PER_BLOCK * DEPTH * DIM];

  const int lane = threadIdx.x & 31;
  const int wv   = __builtin_amdgcn_readfirstlane(threadIdx.x >> 5);  // scalar
  const int node = blockIdx.x * WAVES_PER_BLOCK + wv;                 // scalar
  if (node >= B) return;  // uniform: whole wave exits together

  float* wslot = &smem[wv * DEPTH * DIM];
  // Generic LDS pointers are {SHARED_BASE, offset32}: low 32 bits = DS addr.
  const unsigned lds_lane = (unsigned)(uintptr_t)wslot + (unsigned)(lane * 16);
  const float*   lsrc     = wslot + lane * 4;

  float ax = 0.f, ay = 0.f, az = 0.f, aw = 0.f;
  auto consume = [&](int j) {
    const v4f v = *reinterpret_cast<const v4f*>(lsrc + (j & (DEPTH - 1)) * DIM);
    // Fence: slot must be read before its refill (issued right after) lands.
    wait_dscnt0();
    ax += v.x; ay += v.y; az += v.z; aw += v.w;
  };

  if (K == 32) {
    // ---- specialized, fully-unrolled path (the actual workload) ---------
    // Lane j holds idx[node*32 + j]; broadcast via v_readlane -> SGPR row,
    // scalar base address for the GVS async form.
    const int myidx =
        __builtin_nontemporal_load(neigh_idx + (size_t)node * 32 + lane);
    const uint64_t tbase = (uint64_t)(uintptr_t)table;
    const unsigned voff  = (unsigned)(lane * 16);   // loop-invariant VGPR

    auto issue = [&](int j) {
      const unsigned row = (unsigned)__builtin_amdgcn_readlane(myidx, j);
      async_b128_gvs(lds_lane + (unsigned)((j & (DEPTH - 1)) * ROW_BYTES),
                     voff, tbase + (((uint64_t)row) << 9));
    };

#pragma unroll
    for (int j = 0; j < DEPTH; ++j) issue(j);          // fill pipeline
#pragma unroll
    for (int j = 0; j < 32 - DEPTH; ++j) {             // steady state
      wait_asynccnt<DEPTH - 1>();  // in-order completion -> row j landed
      consume(j);
      issue(j + DEPTH);
    }
    wait_asynccnt<0>();                                // drain
#pragma unroll
    for (int j = 32 - DEPTH; j < 32; ++j) consume(j);
  } else {
    // ---- generic fallback (any K) ---------------------------------------
    int myidx = 0;
    if (lane < K && K <= 32) myidx = neigh_idx[(size_t)node * K + lane];
    const int kBase = node * K;

    auto issue = [&](int j) {
      int row;
      if (K <= 32) row = __shfl(myidx, j);
      else         row = neigh_idx[(size_t)kBase + j];
      const float* gsrc = table + (size_t)(unsigned)row * DIM + lane * 4;
      async_b128_gv(lds_lane + (unsigned)((j & (DEPTH - 1)) * ROW_BYTES), gsrc);
    };

    const int pre = (K < DEPTH) ? K : DEPTH;
    for (int j = 0; j < pre; ++j) issue(j);
    int j = 0;
    for (; j + DEPTH < K; ++j) {
      wait_asynccnt<DEPTH - 1>();
      consume(j);
      issue(j + DEPTH);
    }
    wait_asynccnt<0>();
    for (; j < K; ++j) consume(j);
  }

  const float inv = 1.0f / (float)K;
  v4f r = {ax * inv, ay * inv, az * inv, aw * inv};
  // NT store: write-once output must not evict the L2-resident table.
  __builtin_nontemporal_store(
      r, reinterpret_cast<v4f*>(out + (size_t)node * DIM + lane * 4));
}

extern "C" void kernel_launch(void* const* d_in, const int* in_sizes, int n_in,
                              void* d_out, int out_size, void* d_ws, size_t ws_size,
                              hipStream_t stream) {
  const float* table     = (const float*)d_in[0];   // [V, 128] f32
  const int*   neigh_idx = (const int*)d_in[1];     // [B, K] int
  float*       out       = (float*)d_out;           // [B, 128] f32
  (void)d_ws; (void)ws_size; (void)n_in;

  const int B = out_size / DIM;                 // D = 128 (reference)
  const int K = (B > 0) ? in_sizes[1] / B : 0;  // = 32 (reference)
  if (B <= 0 || K <= 0) return;

  const int blocks = (B + WAVES_PER_BLOCK - 1) / WAVES_PER_BLOCK;
  SpaAggregator_gather_mean_kernel<<<blocks, THREADS_PER_BLOCK, 0, stream>>>(
      table, neigh_idx, out, B, K);
}